// PointGroup_14474039787638
// MI455X (gfx1250) — compile-verified
//
#include <hip/hip_runtime.h>
#include <float.h>

// ---------------------------------------------------------------------------
// PointGroup voxel pooling for MI455X (gfx1250, wave32).
//
//  * Kernel 1: one workgroup per cluster, 1024 threads (32 waves -> 8 per
//    SIMD32, full WGP wave slots for gather latency hiding). All voxel state
//    in LDS (320KB/WGP): point ids, voxel ids, cached coords, voxel counts,
//    and voxel feature accumulators for 16 of 32 channels per pass with a
//    17-word row stride (17 coprime with 64 banks -> conflict-free-ish
//    ds_add_f32). ~237KB LDS total; vfeat (180MB) never touches global.
//  * Kernel 2: scores = pooled(512x32) @ W(32x1) + b via chained
//    V_WMMA_F32_16X16X4_F32 (exact f32), one wave per 16 clusters.
// ---------------------------------------------------------------------------

#define NPTS_TOT  1048576
#define NC        512
#define PPC       2048
#define CCH       32
#define FS        14
#define FS3       (FS*FS*FS)     // 2744
#define VSTRIDE   17             // padded vacc row stride (banks: 17 ⟂ 64)
#define MAX_SCALE 50.0f
#define TPB       1024
#define NWAVES    (TPB/32)

typedef __attribute__((ext_vector_type(2))) float v2f;
typedef __attribute__((ext_vector_type(8))) float v8f;

__global__ void __launch_bounds__(TPB)
cluster_pool_kernel(const float* __restrict__ feats,
                    const float* __restrict__ coords,
                    const float* __restrict__ rand3,
                    const int*   __restrict__ clusters_idx,
                    const int*   __restrict__ clusters_offset,
                    float* __restrict__ pooled)
{
  extern __shared__ float smem[];
  int*   s_pts  = (int*)smem;                 // [PPC]
  int*   s_vid  = s_pts + PPC;                // [PPC]
  float* s_crd  = (float*)(s_vid + PPC);      // [PPC*3] cached coords
  float* s_scr  = s_crd + PPC * 3;            // [1024] reduction scratch
  float* s_vcnt = s_scr + 1024;               // [FS3]
  float* s_vacc = s_vcnt + FS3;               // [FS3*VSTRIDE]

  const int cid  = blockIdx.x;
  const int tid  = threadIdx.x;
  const int lane = tid & 31;
  const int wave = tid >> 5;

  const int start = clusters_offset[cid];
  int npts = clusters_offset[cid + 1] - start;
  if (npts > PPC) npts = PPC;

  // ---- pass 1: gather point ids + coords, coordinate sum/min/max ----------
  float sx = 0.f, sy = 0.f, sz = 0.f;
  float mnx =  FLT_MAX, mny =  FLT_MAX, mnz =  FLT_MAX;
  float mxx = -FLT_MAX, mxy = -FLT_MAX, mxz = -FLT_MAX;
  for (int i = tid; i < npts; i += TPB) {
    int p = clusters_idx[2 * (start + i) + 1];
    s_pts[i] = p;
    float x = coords[3 * (size_t)p + 0];
    float y = coords[3 * (size_t)p + 1];
    float z = coords[3 * (size_t)p + 2];
    s_crd[3 * i + 0] = x;
    s_crd[3 * i + 1] = y;
    s_crd[3 * i + 2] = z;
    sx += x; sy += y; sz += z;
    mnx = fminf(mnx, x); mny = fminf(mny, y); mnz = fminf(mnz, z);
    mxx = fmaxf(mxx, x); mxy = fmaxf(mxy, y); mxz = fmaxf(mxz, z);
  }
  // wave32 butterfly reduction of the 9 partials
  #pragma unroll
  for (int o = 16; o > 0; o >>= 1) {
    sx  += __shfl_xor(sx,  o, 32);
    sy  += __shfl_xor(sy,  o, 32);
    sz  += __shfl_xor(sz,  o, 32);
    mnx  = fminf(mnx, __shfl_xor(mnx, o, 32));
    mny  = fminf(mny, __shfl_xor(mny, o, 32));
    mnz  = fminf(mnz, __shfl_xor(mnz, o, 32));
    mxx  = fmaxf(mxx, __shfl_xor(mxx, o, 32));
    mxy  = fmaxf(mxy, __shfl_xor(mxy, o, 32));
    mxz  = fmaxf(mxz, __shfl_xor(mxz, o, 32));
  }
  if (lane == 0) {
    float* w = s_scr + wave * 9;
    w[0] = sx; w[1] = sy; w[2] = sz;
    w[3] = mnx; w[4] = mny; w[5] = mnz;
    w[6] = mxx; w[7] = mxy; w[8] = mxz;
  }
  __syncthreads();
  sx  = s_scr[0]; sy  = s_scr[1]; sz  = s_scr[2];
  mnx = s_scr[3]; mny = s_scr[4]; mnz = s_scr[5];
  mxx = s_scr[6]; mxy = s_scr[7]; mxz = s_scr[8];
  #pragma unroll
  for (int w = 1; w < NWAVES; ++w) {
    const float* r = s_scr + w * 9;
    sx += r[0]; sy += r[1]; sz += r[2];
    mnx = fminf(mnx, r[3]); mny = fminf(mny, r[4]); mnz = fminf(mnz, r[5]);
    mxx = fmaxf(mxx, r[6]); mxy = fmaxf(mxy, r[7]); mxz = fmaxf(mxz, r[8]);
  }
  __syncthreads();

  // ---- per-cluster transform parameters (computed redundantly) ------------
  const float inv_n = 1.0f / (float)npts;
  const float mex = sx * inv_n, mey = sy * inv_n, mez = sz * inv_n;
  const float cminx = mnx - mex, cminy = mny - mey, cminz = mnz - mez;
  const float cmaxx = mxx - mex, cmaxy = mxy - mey, cmaxz = mxz - mez;
  const float ext = fmaxf(cmaxx - cminx, fmaxf(cmaxy - cminy, cmaxz - cminz))
                    * (1.0f / (float)FS);
  const float cscale = fminf(1.0f / ext - 0.01f, MAX_SCALE);
  const float r0 = rand3[0], r1 = rand3[1], r2 = rand3[2];
  const float minx = cminx * cscale, miny = cminy * cscale, minz = cminz * cscale;
  const float bx = (cmaxx - cminx) * cscale;
  const float by = (cmaxy - cminy) * cscale;
  const float bz = (cmaxz - cminz) * cscale;
  const float ofx = -minx + fmaxf((float)FS - bx - 0.001f, 0.f) * r0
                          + fminf((float)FS - bx + 0.001f, 0.f) * r0;
  const float ofy = -miny + fmaxf((float)FS - by - 0.001f, 0.f) * r1
                          + fminf((float)FS - by + 0.001f, 0.f) * r1;
  const float ofz = -minz + fmaxf((float)FS - bz - 0.001f, 0.f) * r2
                          + fminf((float)FS - bz + 0.001f, 0.f) * r2;

  // ---- pass 2: voxel ids + counts -----------------------------------------
  for (int v = tid; v < FS3; v += TPB) s_vcnt[v] = 0.f;
  __syncthreads();
  for (int i = tid; i < npts; i += TPB) {
    float x = (s_crd[3 * i + 0] - mex) * cscale + ofx;
    float y = (s_crd[3 * i + 1] - mey) * cscale + ofy;
    float z = (s_crd[3 * i + 2] - mez) * cscale + ofz;
    int vx = min(max((int)floorf(x), 0), FS - 1);
    int vy = min(max((int)floorf(y), 0), FS - 1);
    int vz = min(max((int)floorf(z), 0), FS - 1);
    int vid = (vx * FS + vy) * FS + vz;
    s_vid[i] = vid;
    atomicAdd(&s_vcnt[vid], 1.0f);
    // warm L2/WGP$ for the feature-accumulation passes (global_prefetch_b8)
    __builtin_prefetch(&feats[(size_t)s_pts[i] * CCH], 0, 1);
  }
  __syncthreads();

  // ---- pass 3: per-voxel feature mean + per-cluster max, 16 ch at a time --
  #pragma unroll
  for (int h = 0; h < 2; ++h) {
    for (int v = tid; v < FS3 * VSTRIDE; v += TPB) s_vacc[v] = 0.f;
    __syncthreads();
    for (int i = tid; i < npts; i += TPB) {
      int p = s_pts[i];
      const float4* f4 = (const float4*)(feats + (size_t)p * CCH + h * 16);
      int vb = s_vid[i] * VSTRIDE;
      #pragma unroll
      for (int q = 0; q < 4; ++q) {
        float4 f = f4[q];
        atomicAdd(&s_vacc[vb + q * 4 + 0], f.x);
        atomicAdd(&s_vacc[vb + q * 4 + 1], f.y);
        atomicAdd(&s_vacc[vb + q * 4 + 2], f.z);
        atomicAdd(&s_vacc[vb + q * 4 + 3], f.w);
      }
    }
    __syncthreads();
    // max over occupied voxels of (vacc / vcnt)
    float pm[16];
    #pragma unroll
    for (int c = 0; c < 16; ++c) pm[c] = -FLT_MAX;
    for (int v = tid; v < FS3; v += TPB) {
      float cnt = s_vcnt[v];
      if (cnt > 0.f) {
        float inv = 1.0f / cnt;
        #pragma unroll
        for (int c = 0; c < 16; ++c)
          pm[c] = fmaxf(pm[c], s_vacc[v * VSTRIDE + c] * inv);
      }
    }
    #pragma unroll
    for (int c = 0; c < 16; ++c) {
      float v = pm[c];
      #pragma unroll
      for (int o = 16; o > 0; o >>= 1) v = fmaxf(v, __shfl_xor(v, o, 32));
      pm[c] = v;
    }
    __syncthreads();
    if (lane == 0) {
      #pragma unroll
      for (int c = 0; c < 16; ++c) s_scr[wave * 16 + c] = pm[c];
    }
    __syncthreads();
    if (tid < 16) {
      float v = s_scr[tid];
      #pragma unroll
      for (int w = 1; w < NWAVES; ++w) v = fmaxf(v, s_scr[w * 16 + tid]);
      pooled[(size_t)cid * CCH + h * 16 + tid] = v;
    }
    __syncthreads();
  }
}

// ---------------------------------------------------------------------------
// scores = pooled(512x32) @ W(32x1) + b via V_WMMA_F32_16X16X4_F32.
// One wave handles 16 clusters; K=32 accumulated in 8 chained WMMAs of K=4.
// A 16x4 f32 layout (ISA 7.12.2): lanes 0-15 hold {K0,K1}, lanes 16-31 {K2,K3}
// for rows M=lane%16. B 4x16 mirrors it (rows K across lane halves, cols N
// across lanes). D 16x16: column 0 lives in lanes 0 (M=0..7) and 16 (M=8..15).
// ---------------------------------------------------------------------------
__global__ void __launch_bounds__(32)
scores_wmma_kernel(const float* __restrict__ pooled,
                   const float* __restrict__ W,
                   const float* __restrict__ bias,
                   float* __restrict__ out)
{
  const int lane  = threadIdx.x;          // 0..31
  const int nrow  = lane & 15;            // A row / B column
  const int kofs  = (lane >> 4) * 2;      // this lane holds K = kofs, kofs+1
  const int mbase = blockIdx.x * 16;

  v8f acc = {};
  #pragma unroll
  for (int kc = 0; kc < 8; ++kc) {
    const float* prow = pooled + (size_t)(mbase + nrow) * CCH + kc * 4 + kofs;
    v2f a, b;
    a.x = prow[0];
    a.y = prow[1];
    float w0 = W[kc * 4 + kofs + 0];
    float w1 = W[kc * 4 + kofs + 1];
    b.x = (nrow == 0) ? w0 : 0.f;         // W occupies column N=0 of B
    b.y = (nrow == 0) ? w1 : 0.f;
    acc = __builtin_amdgcn_wmma_f32_16x16x4_f32(
        /*neg_a=*/false, a, /*neg_b=*/false, b,
        /*c_mod=*/(short)0, acc, /*reuse_a=*/false, /*reuse_b=*/false);
  }
  float bb = bias[0];
  if (nrow == 0) {                        // lanes 0 and 16 hold column 0
    int m0 = mbase + (lane >> 4) * 8;
    #pragma unroll
    for (int g = 0; g < 8; ++g) out[m0 + g] = acc[g] + bb;
  }
}

extern "C" void kernel_launch(void* const* d_in, const int* in_sizes, int n_in,
                              void* d_out, int out_size, void* d_ws, size_t ws_size,
                              hipStream_t stream)
{
  const float* feats           = (const float*)d_in[0];
  const float* coords          = (const float*)d_in[1];
  const float* W               = (const float*)d_in[2];
  const float* b               = (const float*)d_in[3];
  const float* rand3           = (const float*)d_in[4];
  const int*   clusters_idx    = (const int*)d_in[5];
  const int*   clusters_offset = (const int*)d_in[6];

  float* pooled = (float*)d_ws;           // NC*CCH floats = 64KB
  float* out    = (float*)d_out;          // NC floats

  const size_t smem_bytes =
      (size_t)PPC * 2 * sizeof(int)            // s_pts + s_vid
    + (size_t)PPC * 3 * sizeof(float)          // s_crd
    + 1024 * sizeof(float)                     // s_scr
    + (size_t)FS3 * sizeof(float)              // s_vcnt
    + (size_t)FS3 * VSTRIDE * sizeof(float);   // s_vacc  -> ~237KB < 320KB/WGP

  hipLaunchKernelGGL(cluster_pool_kernel, dim3(NC), dim3(TPB), smem_bytes, stream,
                     feats, coords, rand3, clusters_idx, clusters_offset, pooled);
  hipLaunchKernelGGL(scores_wmma_kernel, dim3(NC / 16), dim3(32), 0, stream,
                     pooled, W, b, out);
}